// Decoder_59270548685369
// MI455X (gfx1250) — compile-verified
//
#include <hip/hip_runtime.h>
#include <hip/hip_bf16.h>
#include <stdint.h>

// ---------------------------------------------------------------------------
// Cosine-sim top-k on MI455X (gfx1250), compile-only tuning.
// Bandwidth-bound (1.54 GB @ 23.3 TB/s ~ 66us floor). Dot products run on the
// full-precision f32 matrix pipe: V_WMMA_F32_16X16X4_F32, chained over K=384.
// Software-pipelined with two 8-segment staging buffers so each wave keeps
// ~8-16 b64 loads in flight (needed to cover HBM latency) while staying well
// under 256 VGPRs for multi-wave occupancy.
// ---------------------------------------------------------------------------

typedef float v2f __attribute__((ext_vector_type(2)));
typedef float v8f __attribute__((ext_vector_type(8)));

#define DIM            384
#define SEGS           (DIM / 4)        // 96 WMMA K-steps per 16-row tile
#define BLOCK_THREADS  256
#define NUM_BLOCKS     512
#define TOTAL_WAVES    (NUM_BLOCKS * BLOCK_THREADS / 32)   // 4096
#define TOPK           5
#define CANDS_PER_WAVE (2 * TOPK)       // lanes 0 and 16 each contribute 5

__device__ __forceinline__ bool better(float av, int ai, float bv, int bi) {
    // (value desc, index asc) — matches jax.lax.top_k tie-breaking
    return (av > bv) || (av == bv && ai < bi);
}

// Issue 8 A-operand loads (one 16-byte K-chunk per lane pair) as one clause.
__device__ __forceinline__ void load8(v2f (&buf)[8],
                                      const float* __restrict__ rowPtr,
                                      int baseT) {
#pragma unroll
    for (int u = 0; u < 8; ++u)
        buf[u] = __builtin_nontemporal_load(
                     (const v2f*)(rowPtr + 4 * (baseT + u)));
}

// Consume 8 staged A operands: 8 WMMAs (4 independent chains) + norm FMAs.
__device__ __forceinline__ void consume8(const v2f (&buf)[8],
                                         const float* qbase,  // LDS q-hat
                                         int baseT, int g2, float& n2,
                                         v8f& acc0, v8f& acc1,
                                         v8f& acc2, v8f& acc3) {
#pragma unroll
    for (int u = 0; u < 8; ++u) {
        const v2f a = buf[u];
        const v2f b = *(const v2f*)(qbase + 4 * (baseT + u) + g2);
        n2 = fmaf(a.x, a.x, fmaf(a.y, a.y, n2));
        const int c = u & 3;
        if (c == 0)
            acc0 = __builtin_amdgcn_wmma_f32_16x16x4_f32(
                false, a, false, b, (short)0, acc0, false, false);
        else if (c == 1)
            acc1 = __builtin_amdgcn_wmma_f32_16x16x4_f32(
                false, a, false, b, (short)0, acc1, false, false);
        else if (c == 2)
            acc2 = __builtin_amdgcn_wmma_f32_16x16x4_f32(
                false, a, false, b, (short)0, acc2, false, false);
        else
            acc3 = __builtin_amdgcn_wmma_f32_16x16x4_f32(
                false, a, false, b, (short)0, acc3, false, false);
    }
}

__global__ __launch_bounds__(BLOCK_THREADS)
void sim_topk_kernel(const float* __restrict__ emb,
                     const float* __restrict__ corpus,
                     int nRows,
                     float* __restrict__ wsVals,
                     int*   __restrict__ wsIdx)
{
    __shared__ float qhat[DIM];
    __shared__ float red[BLOCK_THREADS];

    const int tid = threadIdx.x;

    // ---- normalize query into LDS (per block; 384 floats, negligible) ----
    float s = 0.0f;
    for (int i = tid; i < DIM; i += BLOCK_THREADS) {
        float v = emb[i];
        s = fmaf(v, v, s);
    }
    red[tid] = s;
    __syncthreads();
    for (int off = BLOCK_THREADS / 2; off > 0; off >>= 1) {
        if (tid < off) red[tid] += red[tid + off];
        __syncthreads();
    }
    const float qscale = 1.0f / fmaxf(sqrtf(red[0]), 1e-12f);
    for (int i = tid; i < DIM; i += BLOCK_THREADS) qhat[i] = emb[i] * qscale;
    __syncthreads();

    // ---- per-wave identity ----
    const int lane = tid & 31;
    const int g    = lane >> 4;     // half-wave: 0 => K pair {0,1}, 1 => {2,3}
    const int g2   = 2 * g;
    const int r16  = lane & 15;     // row within 16-row tile
    const int waveGlobal = (blockIdx.x * BLOCK_THREADS + tid) >> 5;
    const int numTiles   = nRows >> 4;

    float tv[TOPK]; int ti[TOPK];
#pragma unroll
    for (int i = 0; i < TOPK; ++i) { tv[i] = -3.0e38f; ti[i] = 0x7fffffff; }

    for (int tile = waveGlobal; tile < numTiles; tile += TOTAL_WAVES) {
        // Lane L streams row (tile*16 + L&15); half-waves split K pairs.
        const float* rowPtr =
            corpus + (size_t)(tile * 16 + r16) * DIM + g2;

        if (tile + TOTAL_WAVES < numTiles)
            __builtin_prefetch(rowPtr + (size_t)TOTAL_WAVES * 16 * DIM, 0, 0);

        v8f acc0 = {}; v8f acc1 = {}; v8f acc2 = {}; v8f acc3 = {};
        float n2 = 0.0f;

        // ---- software pipeline: double-buffered 8-segment stages ----
        v2f A0[8], A1[8];
        load8(A0, rowPtr, 0);
#pragma unroll 1
        for (int t16 = 0; t16 + 16 < SEGS; t16 += 16) {  // t16 = 0..64
            load8(A1, rowPtr, t16 + 8);
            consume8(A0, qhat, t16, g2, n2, acc0, acc1, acc2, acc3);
            load8(A0, rowPtr, t16 + 16);
            consume8(A1, qhat, t16 + 8, g2, n2, acc0, acc1, acc2, acc3);
        }
        // epilogue: segments [SEGS-16, SEGS) — no loads past the row
        load8(A1, rowPtr, SEGS - 8);
        consume8(A0, qhat, SEGS - 16, g2, n2, acc0, acc1, acc2, acc3);
        consume8(A1, qhat, SEGS - 8,  g2, n2, acc0, acc1, acc2, acc3);

        // Complete ||row||^2: lane L and L^16 hold complementary K halves.
        const float n2full = n2 + __shfl_xor(n2, 16, 32);
        const v8f dots = acc0 + acc1 + acc2 + acc3;

        // D layout: acc[j] at lane L is row (8*(L>>4) + j); fetch its norm
        // (resident at lane (8g+j)) while EXEC is still all-ones.
        float rnorm[8];
#pragma unroll
        for (int j = 0; j < 8; ++j)
            rnorm[j] = __shfl(n2full, 8 * g + j, 32);

#pragma unroll
        for (int j = 0; j < 8; ++j) {
            float sim = dots[j] / fmaxf(sqrtf(rnorm[j]), 1e-12f);
            sim = (sim != sim) ? 0.0f : fmaxf(sim, 1e-6f);  // nan_to_num+clamp
            const int row = tile * 16 + 8 * g + j;
            if (r16 == 0) {  // dedupe: columns replicate; lanes 0/16 own rows
                if (better(sim, row, tv[TOPK - 1], ti[TOPK - 1])) {
                    tv[TOPK - 1] = sim; ti[TOPK - 1] = row;
#pragma unroll
                    for (int p = TOPK - 1; p > 0; --p) {
                        if (better(tv[p], ti[p], tv[p - 1], ti[p - 1])) {
                            float fv = tv[p]; tv[p] = tv[p - 1]; tv[p - 1] = fv;
                            int   iv = ti[p]; ti[p] = ti[p - 1]; ti[p - 1] = iv;
                        }
                    }
                }
            }
        }
    }

    // ---- spill per-wave candidates (every slot written every call) ----
    if (r16 == 0) {
        float* wv = wsVals + (size_t)waveGlobal * CANDS_PER_WAVE + g * TOPK;
        int*   wi = wsIdx  + (size_t)waveGlobal * CANDS_PER_WAVE + g * TOPK;
#pragma unroll
        for (int i = 0; i < TOPK; ++i) { wv[i] = tv[i]; wi[i] = ti[i]; }
    }
}

__global__ __launch_bounds__(256)
void topk_reduce_kernel(const float* __restrict__ wsVals,
                        const int*   __restrict__ wsIdx,
                        int nCand,
                        float* __restrict__ out)
{
    __shared__ float sv[256 * TOPK];
    __shared__ int   si[256 * TOPK];
    const int tid = threadIdx.x;

    float tv[TOPK]; int ti[TOPK];
#pragma unroll
    for (int i = 0; i < TOPK; ++i) { tv[i] = -3.0e38f; ti[i] = 0x7fffffff; }

    for (int i = tid; i < nCand; i += 256) {
        const float v = wsVals[i];
        const int   r = wsIdx[i];
        if (better(v, r, tv[TOPK - 1], ti[TOPK - 1])) {
            tv[TOPK - 1] = v; ti[TOPK - 1] = r;
#pragma unroll
            for (int p = TOPK - 1; p > 0; --p) {
                if (better(tv[p], ti[p], tv[p - 1], ti[p - 1])) {
                    float fv = tv[p]; tv[p] = tv[p - 1]; tv[p - 1] = fv;
                    int   iv = ti[p]; ti[p] = ti[p - 1]; ti[p - 1] = iv;
                }
            }
        }
    }
#pragma unroll
    for (int i = 0; i < TOPK; ++i) { sv[tid * TOPK + i] = tv[i]; si[tid * TOPK + i] = ti[i]; }
    __syncthreads();

    if (tid == 0) {
        float fv_[TOPK]; int fi_[TOPK];
#pragma unroll
        for (int i = 0; i < TOPK; ++i) { fv_[i] = -3.0e38f; fi_[i] = 0x7fffffff; }
        for (int i = 0; i < 256 * TOPK; ++i) {
            const float v = sv[i]; const int r = si[i];
            if (better(v, r, fv_[TOPK - 1], fi_[TOPK - 1])) {
                fv_[TOPK - 1] = v; fi_[TOPK - 1] = r;
#pragma unroll
                for (int p = TOPK - 1; p > 0; --p) {
                    if (better(fv_[p], fi_[p], fv_[p - 1], fi_[p - 1])) {
                        float a = fv_[p]; fv_[p] = fv_[p - 1]; fv_[p - 1] = a;
                        int   b = fi_[p]; fi_[p] = fi_[p - 1]; fi_[p - 1] = b;
                    }
                }
            }
        }
#pragma unroll
        for (int i = 0; i < TOPK; ++i) {
            out[i]        = fv_[i];          // top-5 similarities
            out[TOPK + i] = (float)fi_[i];   // top-5 indices (f32 output buf)
        }
    }
}

extern "C" void kernel_launch(void* const* d_in, const int* in_sizes, int n_in,
                              void* d_out, int out_size, void* d_ws, size_t ws_size,
                              hipStream_t stream) {
    const float* emb    = (const float*)d_in[0];
    const float* corpus = (const float*)d_in[1];
    const int nRows = in_sizes[1] / DIM;

    const int nCand = TOTAL_WAVES * CANDS_PER_WAVE;          // 40960
    float* wsVals = (float*)d_ws;                            // 160 KB
    int*   wsIdx  = (int*)((char*)d_ws + (size_t)nCand * sizeof(float));

    sim_topk_kernel<<<NUM_BLOCKS, BLOCK_THREADS, 0, stream>>>(
        emb, corpus, nRows, wsVals, wsIdx);
    topk_reduce_kernel<<<1, 256, 0, stream>>>(
        wsVals, wsIdx, nCand, (float*)d_out);
}